// BlockDWT2d_12378095747848
// MI455X (gfx1250) — compile-verified
//
#include <hip/hip_runtime.h>

typedef float v2f __attribute__((ext_vector_type(2)));
typedef float v8f __attribute__((ext_vector_type(8)));

__device__ __forceinline__ float hsign(int u, int p) {
    // natural-order Hadamard entry: (-1)^popcount(u&p)
    return (__builtin_popcount(u & p) & 1) ? -1.0f : 1.0f;
}

// One workgroup = one 16-row strip of one (n,c) 512x512 plane.
// 8 waves, each handling 4 tiles of 16x16 input (32 tiles across the strip).
// Per tile: Y = (1/8) * Hb * X * Hb, Hb = blockdiag(H8, H8), via
// 8x v_wmma_f32_16x16x4_f32 (two 16x16x16 matmuls, K in chunks of 4).
__global__ __launch_bounds__(256) void haar3_wmma_kernel(const float* __restrict__ in,
                                                         float* __restrict__ out) {
    __shared__ float zs[8 * 256];     // per-wave 16x16 intermediate Z
    __shared__ float obuf[64 * 128];  // staged output: [chIdx(64)][a(2)][col(64)]

    const int tid  = threadIdx.x;
    const int lane = tid & 31;
    const int wave = tid >> 5;
    const int hf   = lane >> 4;   // half-wave index (K/M split)
    const int col  = lane & 15;

    const int rg = blockIdx.x;    // row group: input rows [16*rg, 16*rg+16)
    const int c  = blockIdx.y;    // color channel 0..2
    const int n  = blockIdx.z;    // batch 0..31

    const float* plane = in + ((size_t)(n * 3 + c)) * (512 * 512) + (size_t)rg * 16 * 512;

    // Hb chunks. By symmetry of Hb, the per-lane A-layout values (row=col, K=k0..k0+1)
    // equal the B-layout values (rows k0..k0+1, col=col). One array serves both.
    v2f hb[4];
#pragma unroll
    for (int q = 0; q < 4; ++q) {
        const int k0 = q * 4 + 2 * hf;
        const int m  = col;
        hb[q].x = (((m ^ k0) & 8) ? 0.0f : hsign(m & 7, k0 & 7));
        hb[q].y = (((m ^ (k0 + 1)) & 8) ? 0.0f : hsign(m & 7, (k0 + 1) & 7));
    }

    float* zw = zs + wave * 256;

    // per-lane output-channel pieces from the column-transform index v = N&7
    const int v     = col & 7;
    const int bb    = col >> 3;                                        // block col within tile
    const int vPart = (v & 1) + ((v >> 1) & 1) * 4 + ((v >> 2) & 1) * 16;

    for (int k = 0; k < 4; ++k) {
        const int t = wave + 8 * k;              // tile column 0..31
        const float* tsrc = plane + t * 16;

        if (k < 3) {
            // prefetch next tile's rows (CDNA5 global_prefetch_b8)
            __builtin_prefetch(tsrc + 128 + col * 512, 0, 3);
        }

        // ---- Z = Hb * X : A = Hb (regs), B = X (global, B layout) ----
        v8f zacc = {0.f, 0.f, 0.f, 0.f, 0.f, 0.f, 0.f, 0.f};
#pragma unroll
        for (int q = 0; q < 4; ++q) {
            const int r0 = q * 4 + 2 * hf;
            v2f bx;
            bx.x = tsrc[r0 * 512 + col];
            bx.y = tsrc[(r0 + 1) * 512 + col];
            zacc = __builtin_amdgcn_wmma_f32_16x16x4_f32(
                false, hb[q], false, bx, (short)0, zacc, false, false);
        }

        // spill Z (D layout) to LDS so it can be re-read in A layout
#pragma unroll
        for (int j = 0; j < 8; ++j) {
            zw[(j + 8 * hf) * 16 + col] = zacc[j];
        }
        __asm__ volatile("s_wait_dscnt 0x0" ::: "memory");

        // ---- Y = Z * Hb : A = Z (LDS, A layout), B = Hb (regs) ----
        v8f yacc = {0.f, 0.f, 0.f, 0.f, 0.f, 0.f, 0.f, 0.f};
#pragma unroll
        for (int q = 0; q < 4; ++q) {
            const int k0 = q * 4 + 2 * hf;
            v2f av;
            av.x = zw[col * 16 + k0];
            av.y = zw[col * 16 + k0 + 1];
            yacc = __builtin_amdgcn_wmma_f32_16x16x4_f32(
                false, av, false, hb[q], (short)0, yacc, false, false);
        }
        __asm__ volatile("s_wait_dscnt 0x0" ::: "memory");

        // ---- stage scaled coefficients into output LDS ----
        // D layout: VGPR j holds Y[M = j + 8*hf][N = col]; u = M&7 = j, a = hf.
        const int base = vPart * 128 + hf * 64 + 2 * t + bb;
#pragma unroll
        for (int j = 0; j < 8; ++j) {
            const int uPart = (j & 1) * 2 + ((j >> 1) & 1) * 8 + ((j >> 2) & 1) * 32;
            obuf[uPart * 128 + base] = yacc[j] * 0.125f;
        }
    }

    __syncthreads();

    // ---- coalesced store: per chIdx, rows (2*rg, 2*rg+1) x 64 cols are
    // 128 contiguous floats in the 64x64 output plane ----
    float* oplane = out + ((size_t)n * 192 + c) * 4096 + (size_t)rg * 128;
#pragma unroll
    for (int f = tid; f < 2048; f += 256) {
        const int ch = f >> 5;
        const int q  = f & 31;
        const float4 val = *(const float4*)(obuf + ch * 128 + q * 4);
        *(float4*)(oplane + (size_t)(3 * ch) * 4096 + q * 4) = val;
    }
}

extern "C" void kernel_launch(void* const* d_in, const int* in_sizes, int n_in,
                              void* d_out, int out_size, void* d_ws, size_t ws_size,
                              hipStream_t stream) {
    (void)in_sizes; (void)n_in; (void)d_ws; (void)ws_size; (void)out_size;
    const float* x = (const float*)d_in[0];
    float* out = (float*)d_out;
    dim3 grid(32, 3, 32);   // (row-group, c, n)
    dim3 block(256);
    hipLaunchKernelGGL(haar3_wmma_kernel, grid, block, 0, stream, x, out);
}